// VectorFitter_36610301231223
// MI455X (gfx1250) — compile-verified
//
#include <hip/hip_runtime.h>
#include <math.h>

typedef float v2f __attribute__((ext_vector_type(2)));
typedef float v8f __attribute__((ext_vector_type(8)));

namespace {
constexpr int   NP      = 512;
constexpr float DXY     = 2.0f / 512.0f;
constexpr float KMED    = (float)((1.49 * 1.49) / (1.33 * 1.33));    // NA^2/refmed^2
constexpr float KCOV    = (float)((1.49 * 1.49) / (1.52 * 1.52));    // NA^2/refcov^2
constexpr float SQKMED  = (float)(1.49 / 1.33);                      // NA/refmed = sqrt(KMED)
constexpr float REFMED  = 1.33f;
constexpr float REFCOV  = 1.52f;
constexpr float REFIMM  = 1.52f;
constexpr float PHASE_K = (float)(6.283185307179586 / 640.0);        // 2*pi/lambda
constexpr float KXY     = (float)(6.283185307179586 * 1.49 / 640.0); // 2*pi*NA/lambda
constexpr float KZ      = (float)(6.283185307179586 * 1.33 / 640.0); // 2*pi*refmed/lambda
constexpr int   OFF_W   = NP * NP * 12;          // float offset of W block
constexpr int   OFF_WV  = OFF_W + NP * NP * 2;   // float offset of wavevector block
}

struct cpx { float re, im; };
__device__ __forceinline__ cpx cmul(cpx a, cpx b) { return { a.re*b.re - a.im*b.im, a.re*b.im + a.im*b.re }; }
__device__ __forceinline__ cpx cadd(cpx a, cpx b) { return { a.re + b.re, a.im + b.im }; }
__device__ __forceinline__ cpx csub(cpx a, cpx b) { return { a.re - b.re, a.im - b.im }; }
__device__ __forceinline__ cpx cscale(cpx a, float s) { return { a.re * s, a.im * s }; }
__device__ __forceinline__ cpx cdiv(cpx a, cpx b) {
  float inv = 1.0f / (b.re*b.re + b.im*b.im);
  return { (a.re*b.re + a.im*b.im) * inv, (a.im*b.re - a.re*b.im) * inv };
}

__device__ __forceinline__ float grid_coord(int k) { return -1.0f + ((float)k + 0.5f) * DXY; }

// All 12 Zernike terms as pure polynomials in (x, y):
//   rho^2 sin2p = 2xy, rho^2 cos2p = x^2-y^2, rho^3 sin3p = y(3x^2-y^2),
//   rho^3 cos3p = x(x^2-3y^2), R(rho)*rho*{sin,cos}p = R(rho)*{y,x}.
// Radial recursion coefficients are faithful to the reference.
struct Z12 { float v0,v1,v2,v3,v4,v5,v6,v7,v8,v9,v10,v11; };
__device__ __forceinline__ Z12 zern(float x, float y) {
  float x2 = x*x, y2 = y*y;
  float r2 = x2 + y2;
  float t  = 2.0f*r2 - 1.0f;
  float z20 = t;                                               // R_2^0
  float z40 = (48.0f*t*z20 - 16.0f) * (1.0f/32.0f);            // R_4^0
  float z60 = (240.0f*t*z40 - 96.0f*z20) * (1.0f/144.0f);      // R_6^0
  float g3  = 3.0f*r2 - 2.0f;                                  // R_3^1 / rho
  float g5  = (8.0f*(15.0f*t - 1.0f)*g3 - 40.0f)*(1.0f/72.0f); // R_5^1 / rho
  float g4  = 4.0f*r2 - 3.0f;                                  // R_4^2 / rho^2
  Z12 Z;
  Z.v0  = 2.0f*x*y;            // (2,-2)
  Z.v1  = x2 - y2;             // (2, 2)
  Z.v2  = g3*y;                // (3,-1)
  Z.v3  = g3*x;                // (3, 1)
  Z.v4  = z40;                 // (4, 0)
  Z.v5  = y*(3.0f*x2 - y2);    // (3,-3)
  Z.v6  = x*(x2 - 3.0f*y2);    // (3, 3)
  Z.v7  = g4*(2.0f*x*y);       // (4,-2)
  Z.v8  = g4*(x2 - y2);        // (4, 2)
  Z.v9  = g5*y;                // (5,-1)
  Z.v10 = g5*x;                // (5, 1)
  Z.v11 = z60;                 // (6, 0)
  return Z;
}

__device__ __forceinline__ float pick8(const v8f& a, int s) {
  float a01 = (s & 1) ? a[1] : a[0];
  float a23 = (s & 1) ? a[3] : a[2];
  float a45 = (s & 1) ? a[5] : a[4];
  float a67 = (s & 1) ? a[7] : a[6];
  float a03 = (s & 2) ? a23 : a01;
  float a47 = (s & 2) ? a67 : a45;
  return (s & 4) ? a47 : a03;
}

__global__ __launch_bounds__(256)
void pupil_matrix_kernel(const float* __restrict__ aberr,
                         const float* __restrict__ zvals,
                         float* __restrict__ out)
{
  const int lane    = threadIdx.x & 31;
  const int gwave   = (blockIdx.x << 3) + (threadIdx.x >> 5);
  const int base    = gwave << 5;            // 32 pixels per wave, all in one row
  const int colbase = base & (NP - 1);
  const float x     = grid_coord(base >> 9); // row coordinate: wave-uniform

  // coefficients: normfac * aberrations[:,2] — named scalars (no indexable arrays)
  const float c0  = 2.4494897f * aberr[2];
  const float c1  = 2.4494897f * aberr[5];
  const float c2  = 2.8284271f * aberr[8];
  const float c3  = 2.8284271f * aberr[11];
  const float c4  = 2.2360680f * aberr[14];
  const float c5  = 2.8284271f * aberr[17];
  const float c6  = 2.8284271f * aberr[20];
  const float c7  = 3.1622777f * aberr[23];
  const float c8  = 3.1622777f * aberr[26];
  const float c9  = 3.4641016f * aberr[29];
  const float c10 = 3.4641016f * aberr[32];
  const float c11 = 2.6457513f * aberr[35];
  const float zv0 = zvals[0], zv1 = zvals[1], zv2 = zvals[2];

  // ------- Zernike contraction: 2 tiles x 3 chained V_WMMA_F32_16X16X4_F32 -------
  // A layout (16x4 f32): lanes 0-15 supply K={0,1} of row M=lane; lanes 16-31 supply
  // K={2,3} of row M=lane-16. B carries coefficients replicated across all columns.
  const int  mA  = lane & 15;
  const bool hiK = (lane >= 16);
  const Z12 ZA = zern(x, grid_coord(colbase + mA));        // tile0: pixels base+0..15
  const Z12 ZB = zern(x, grid_coord(colbase + 16 + mA));   // tile1: pixels base+16..31

  const float b0x = hiK ? c2  : c0,  b0y = hiK ? c3  : c1;
  const float b1x = hiK ? c6  : c4,  b1y = hiK ? c7  : c5;
  const float b2x = hiK ? c10 : c8,  b2y = hiK ? c11 : c9;

  v8f acc0 = {0.f,0.f,0.f,0.f,0.f,0.f,0.f,0.f};
  v8f acc1 = {0.f,0.f,0.f,0.f,0.f,0.f,0.f,0.f};
  {
    v2f a, b;
    b[0] = b0x; b[1] = b0y;
    a[0] = hiK ? ZA.v2 : ZA.v0;  a[1] = hiK ? ZA.v3 : ZA.v1;
    acc0 = __builtin_amdgcn_wmma_f32_16x16x4_f32(false, a, false, b, (short)0, acc0, false, false);
    a[0] = hiK ? ZB.v2 : ZB.v0;  a[1] = hiK ? ZB.v3 : ZB.v1;
    acc1 = __builtin_amdgcn_wmma_f32_16x16x4_f32(false, a, false, b, (short)0, acc1, false, false);

    b[0] = b1x; b[1] = b1y;
    a[0] = hiK ? ZA.v6 : ZA.v4;  a[1] = hiK ? ZA.v7 : ZA.v5;
    acc0 = __builtin_amdgcn_wmma_f32_16x16x4_f32(false, a, false, b, (short)0, acc0, false, false);
    a[0] = hiK ? ZB.v6 : ZB.v4;  a[1] = hiK ? ZB.v7 : ZB.v5;
    acc1 = __builtin_amdgcn_wmma_f32_16x16x4_f32(false, a, false, b, (short)0, acc1, false, false);

    b[0] = b2x; b[1] = b2y;
    a[0] = hiK ? ZA.v10 : ZA.v8;  a[1] = hiK ? ZA.v11 : ZA.v9;
    acc0 = __builtin_amdgcn_wmma_f32_16x16x4_f32(false, a, false, b, (short)0, acc0, false, false);
    a[0] = hiK ? ZB.v10 : ZB.v8;  a[1] = hiK ? ZB.v11 : ZB.v9;
    acc1 = __builtin_amdgcn_wmma_f32_16x16x4_f32(false, a, false, b, (short)0, acc1, false, false);
  }

  // D layout: lanes 0-15 hold M=0..7 in acc[0..7]; lanes 16-31 hold M=8..15.
  // Lane -> distinct output pixel: tile = (lane>>3)&1, row-half = (lane>>4)&1.
  const int s    = lane & 7;
  const int sel8 = (lane >> 3) & 1;
  const int hi16 = (lane >> 4) & 1;
  const int off  = sel8 * 16 + hi16 * 8 + s;
  const int po   = base + off;
  const float Wz = sel8 ? pick8(acc1, s) : pick8(acc0, s);

  // ------------------------- per-pixel physics -------------------------
  const float y      = grid_coord(colbase + off);
  const float rhosq  = x*x + y*y;
  const bool  inmask = rhosq < 1.0f;

  const float argMed = 1.0f - rhosq * KMED;
  const float sMed   = sqrtf(fabsf(argMed));
  const cpx CTmed = (argMed >= 0.0f) ? cpx{sMed, 0.0f} : cpx{0.0f, -sMed};   // -i branch

  const float argCov = 1.0f - rhosq * KCOV;
  const float sCov   = sqrtf(fabsf(argCov));
  const cpx CTcov = (argCov >= 0.0f) ? cpx{sCov, 0.0f} : cpx{0.0f, sCov};    // principal sqrt
  // CosThetaImm == CosThetaImmnom == CosThetaCov (all n = 1.52)

  // Fresnel; the cov->imm factors are exactly 2nC/(2nC) == 1 inside the mask
  const cpx numMed = cscale(CTmed, 2.0f * REFMED);
  const cpx FP = cdiv(numMed, cadd(cscale(CTcov, REFMED), cscale(CTmed, REFCOV)));
  const cpx FS = cdiv(numMed, cadd(cscale(CTmed, REFMED), cscale(CTcov, REFCOV)));

  // CosPhi^2 = x^2/r2 etc.; SinTheta*CosPhi = (NA/nmed)*x, SinTheta*SinPhi = (NA/nmed)*y
  const float ir2 = 1.0f / rhosq;
  const float cc  = x*x*ir2, ss = y*y*ir2, sc = x*y*ir2;

  const cpx FPC = cmul(FP, CTmed);
  const cpx P00 = cadd(cscale(FPC, cc), cscale(FS, ss));
  const cpx P01 = cscale(csub(FPC, FS), sc);
  const cpx P02 = cscale(FP, -SQKMED * x);
  const cpx P11 = cadd(cscale(FPC, ss), cscale(FS, cc));
  const cpx P12 = cscale(FP, -SQKMED * y);
  // P10 == P01

  // Amplitude = sqrt(CosThetaImm) / (refmed*CosThetaMed); CTcov real > 0 inside mask
  const cpx Amp = cdiv(cpx{sqrtf(sCov), 0.0f}, cscale(CTmed, REFMED));

  // W = Zernike sum + defocus terms
  cpx W = {Wz, 0.0f};
  W = cadd(W, cscale(CTcov, zv0 * REFIMM));
  W = csub(W, cscale(CTcov, zv1 * REFIMM));
  W = csub(W, cscale(CTmed, zv2 * REFMED));

  float ps, pc;
  sincosf(PHASE_K * W.re, &ps, &pc);
  const cpx AP = cmul(Amp, cpx{pc, ps});

  const cpx e00 = inmask ? cmul(AP, P00) : cpx{0.f, 0.f};
  const cpx e01 = inmask ? cmul(AP, P01) : cpx{0.f, 0.f};
  const cpx e02 = inmask ? cmul(AP, P02) : cpx{0.f, 0.f};
  const cpx e11 = inmask ? cmul(AP, P11) : cpx{0.f, 0.f};
  const cpx e12 = inmask ? cmul(AP, P12) : cpx{0.f, 0.f};

  float4* pm = reinterpret_cast<float4*>(out + (size_t)po * 12);
  pm[0] = make_float4(e00.re, e00.im, e01.re, e01.im);
  pm[1] = make_float4(e02.re, e02.im, e01.re, e01.im);   // P10 == P01
  pm[2] = make_float4(e11.re, e11.im, e12.re, e12.im);

  float2* wout = reinterpret_cast<float2*>(out + OFF_W) + po;
  *wout = inmask ? make_float2(W.re, W.im) : make_float2(0.0f, 0.0f);

  float2* wv = reinterpret_cast<float2*>(out + OFF_WV + (size_t)po * 6);
  wv[0] = make_float2(KXY * x, 0.0f);
  wv[1] = make_float2(KXY * y, 0.0f);
  wv[2] = make_float2(KZ * CTmed.re, KZ * CTmed.im);
}

extern "C" void kernel_launch(void* const* d_in, const int* in_sizes, int n_in,
                              void* d_out, int out_size, void* d_ws, size_t ws_size,
                              hipStream_t stream) {
  const float* aberr = (const float*)d_in[0];   // (12,3) float32
  const float* zvals = (const float*)d_in[1];   // (3,)  float32
  float* out = (float*)d_out;
  // 512*512 px, 32 per wave, 8 waves per 256-thread block -> 1024 blocks
  pupil_matrix_kernel<<<dim3(1024), dim3(256), 0, stream>>>(aberr, zvals, out);
}